// GCN_20985210208423
// MI455X (gfx1250) — compile-verified
//
#include <hip/hip_runtime.h>
#include <math.h>
#include <stdint.h>

typedef __attribute__((ext_vector_type(2))) float v2f;
typedef __attribute__((ext_vector_type(8))) float v8f;

// ---------------------------------------------------------------------------
// Degree accumulation: one thread per edge, hardware f32 atomics.
// ---------------------------------------------------------------------------
__global__ __launch_bounds__(256)
void gcn_degree(const int* __restrict__ src, const int* __restrict__ dst,
                float* __restrict__ deg_out, float* __restrict__ deg_in, int E) {
    int i = blockIdx.x * 256 + threadIdx.x;
    if (i < E) {
        unsafeAtomicAdd(&deg_out[src[i]], 1.0f);
        unsafeAtomicAdd(&deg_in[dst[i]], 1.0f);
    }
}

__global__ __launch_bounds__(256)
void gcn_deginv(const float* __restrict__ deg_out, const float* __restrict__ deg_in,
                float* __restrict__ dinv_out, float* __restrict__ dinv_in, int N) {
    int i = blockIdx.x * 256 + threadIdx.x;
    if (i < N) {
        float a = deg_out[i]; a = (a < 1.0f) ? 1.0f : a;
        float b = deg_in[i];  b = (b < 1.0f) ? 1.0f : b;
        dinv_out[i] = rsqrtf(a);
        dinv_in[i]  = rsqrtf(b);
    }
}

// h = x * dout^{-1/2} (row broadcast), float4 vectorized (N*32 float4s).
__global__ __launch_bounds__(256)
void gcn_scale(const float* __restrict__ x, const float* __restrict__ dinv_out,
               float* __restrict__ h, int n4) {
    int i = blockIdx.x * 256 + threadIdx.x;
    if (i < n4) {
        int row = i >> 5;
        float d = dinv_out[row];
        float4 v = reinterpret_cast<const float4*>(x)[i];
        v.x *= d; v.y *= d; v.z *= d; v.w *= d;
        reinterpret_cast<float4*>(h)[i] = v;
    }
}

// ---------------------------------------------------------------------------
// SpMM: one wave (32 lanes) per edge; each lane moves a float4 (512B/edge).
// Gather h[src] (L2-resident: N*128*4B = 51 MB << 192 MB L2) and scatter-add
// to agg[dst] with global_atomic_add_f32 (no-return form).
// ---------------------------------------------------------------------------
__global__ __launch_bounds__(256)
void gcn_spmm(const float* __restrict__ h, const int* __restrict__ src,
              const int* __restrict__ dst, float* __restrict__ agg, int E) {
    const int wid  = (blockIdx.x * 256 + threadIdx.x) >> 5;
    const int lane = threadIdx.x & 31;
    if (wid >= E) return;
    const int s = src[wid];
    const int d = dst[wid];
    const float4 v = *reinterpret_cast<const float4*>(h + (size_t)s * 128 + lane * 4);
    float* p = agg + (size_t)d * 128 + lane * 4;
    unsafeAtomicAdd(p + 0, v.x);
    unsafeAtomicAdd(p + 1, v.y);
    unsafeAtomicAdd(p + 2, v.z);
    unsafeAtomicAdd(p + 3, v.w);
}

// ---------------------------------------------------------------------------
// Dense GEMM: Out[m, n] = (agg[m,:] * din[m]) @ W[:, n] + bias[n]
// via V_WMMA_F32_16X16X4_F32. 256 threads = 8 waves; each wave owns a
// 16-row x NCOLS tile; block covers 128 rows.
// W staged in LDS in K-pair-major float2 layout so each B fragment is one
// aligned ds_load_b64 directly into the WMMA source VGPR pair (no shuffles).
// Bias staged via gfx1250 GLOBAL_LOAD_ASYNC_TO_LDS_B32 (ASYNCcnt path).
// Optionally fuses BN sum / sum-of-squares (ds_add_f32 reduction, then one
// global atomic per feature per block).
// ---------------------------------------------------------------------------
template <int NCOLS, bool STATS>
__global__ __launch_bounds__(256)
void gcn_gemm(const float* __restrict__ A, const float* __restrict__ dinv_in,
              const float* __restrict__ W, const float* __restrict__ bias,
              float* __restrict__ Out, int out_ld, int ncols_real,
              float* __restrict__ bn_sum, float* __restrict__ bn_ssq, int N) {
    constexpr int NT = NCOLS / 16;
    __shared__ float2 lwp[64 * NCOLS];  // pair p -> (W[2p][n], W[2p+1][n])
    __shared__ float  lbias[NCOLS];
    __shared__ float  bs[NCOLS];
    __shared__ float  bq[NCOLS];

    const int tid = threadIdx.x;

    // Zero LDS (covers zero-padding for the C=40 case) and stat slots.
    float* lwf = reinterpret_cast<float*>(lwp);
    for (int i = tid; i < 128 * NCOLS; i += 256) lwf[i] = 0.0f;
    for (int i = tid; i < NCOLS; i += 256) {
        lbias[i] = 0.0f;
        if constexpr (STATS) { bs[i] = 0.0f; bq[i] = 0.0f; }
    }
    __syncthreads();

    // Async-stage bias into LDS: global_load_async_to_lds_b32 (ASYNCcnt).
    if (tid < ncols_real) {
        unsigned lds_off = (unsigned)(uintptr_t)&lbias[tid];
        const float* gaddr = bias + tid;
        asm volatile("global_load_async_to_lds_b32 %0, %1, off"
                     :: "v"(lds_off), "v"(gaddr) : "memory");
    }

    // Stage W (128 x ncols_real row-major in memory) as K-pairs in LDS.
    for (int e = tid; e < 64 * ncols_real; e += 256) {
        int p = e / ncols_real;
        int n = e - p * ncols_real;
        float2 w2;
        w2.x = W[(2 * p)     * ncols_real + n];
        w2.y = W[(2 * p + 1) * ncols_real + n];
        lwp[p * NCOLS + n] = w2;
    }
    asm volatile("s_wait_asynccnt 0x0" ::: "memory");
    __syncthreads();

    const int wave = tid >> 5;
    const int lane = tid & 31;
    const int half = lane >> 4;     // 0: K={0,1}, 1: K={2,3} within a step
    const int l16  = lane & 15;
    const int rowbase = blockIdx.x * 128 + wave * 16;

    int arow = rowbase + l16;
    if (arow > N - 1) arow = N - 1;               // clamp: keep EXEC all-ones
    const float dv = dinv_in[arow];
    const float2* __restrict__ ap =
        reinterpret_cast<const float2*>(A + (size_t)arow * 128);

    v8f acc[NT] = {};

    for (int kk = 0; kk < 32; ++kk) {
        const int p = kk * 2 + half;               // K-pair index
        float2 av = ap[p];                         // A[arow][2p], A[arow][2p+1]
        v2f a;
        a.x = av.x * dv;
        a.y = av.y * dv;
#pragma unroll
        for (int nt = 0; nt < NT; ++nt) {
            float2 bv = lwp[p * NCOLS + nt * 16 + l16];   // single ds_load_b64
            v2f b;
            b.x = bv.x;
            b.y = bv.y;
            acc[nt] = __builtin_amdgcn_wmma_f32_16x16x4_f32(
                false, a, false, b, (short)0, acc[nt], false, false);
        }
    }

    // Epilogue: bias add (LDS, zero-padded), guarded stores, fused BN stats.
#pragma unroll
    for (int nt = 0; nt < NT; ++nt) {
        const int n = nt * 16 + l16;
        const float bcol = lbias[n];
        float s = 0.0f, q = 0.0f;
#pragma unroll
        for (int r = 0; r < 8; ++r) {
            const int m = rowbase + r + half * 8;   // C layout: VGPR r -> M=r / r+8
            float v = acc[nt][r] + bcol;
            if (m < N && n < ncols_real) Out[(size_t)m * out_ld + n] = v;
            if constexpr (STATS) {
                float msk = (m < N) ? 1.0f : 0.0f;
                s += v * msk;
                q += v * v * msk;
            }
        }
        if constexpr (STATS) {
            unsafeAtomicAdd(&bs[n], s);
            unsafeAtomicAdd(&bq[n], q);
        }
    }
    if constexpr (STATS) {
        __syncthreads();
        if (tid < NCOLS) {
            unsafeAtomicAdd(&bn_sum[tid], bs[tid]);
            unsafeAtomicAdd(&bn_ssq[tid], bq[tid]);
        }
    }
}

// Fold BN into per-feature scale/shift: y = v*scale + shift, then ReLU.
__global__ void gcn_bnfinal(const float* __restrict__ bn_sum, const float* __restrict__ bn_ssq,
                            const float* __restrict__ g, const float* __restrict__ be,
                            float* __restrict__ scale, float* __restrict__ shift, float invN) {
    int n = threadIdx.x;   // 128 threads
    float mu  = bn_sum[n] * invN;
    float var = bn_ssq[n] * invN - mu * mu;
    float sc  = g[n] * rsqrtf(var + 1e-5f);
    scale[n] = sc;
    shift[n] = be[n] - mu * sc;
}

// h = relu(o*scale + shift) * dout^{-1/2}  (pre-scaled for next layer's gather)
__global__ __launch_bounds__(256)
void gcn_bnapply(const float* __restrict__ o, const float* __restrict__ scale,
                 const float* __restrict__ shift, const float* __restrict__ dinv_out,
                 float* __restrict__ h, int n4) {
    int i = blockIdx.x * 256 + threadIdx.x;
    if (i < n4) {
        int row = i >> 5;
        int q   = i & 31;
        float4 v  = reinterpret_cast<const float4*>(o)[i];
        float4 sc = reinterpret_cast<const float4*>(scale)[q];
        float4 sh = reinterpret_cast<const float4*>(shift)[q];
        float d = dinv_out[row];
        v.x = fmaxf(v.x * sc.x + sh.x, 0.0f) * d;
        v.y = fmaxf(v.y * sc.y + sh.y, 0.0f) * d;
        v.z = fmaxf(v.z * sc.z + sh.z, 0.0f) * d;
        v.w = fmaxf(v.w * sc.w + sh.w, 0.0f) * d;
        reinterpret_cast<float4*>(h)[i] = v;
    }
}

// In-place row-wise log_softmax over C classes; one thread per row.
__global__ __launch_bounds__(256)
void gcn_logsoftmax(float* __restrict__ out, int N, int C) {
    int row = blockIdx.x * 256 + threadIdx.x;
    if (row >= N) return;
    float v[64];
    float mx = -3.402823466e+38f;
    for (int j = 0; j < C; ++j) {
        v[j] = out[(size_t)row * C + j];
        mx = fmaxf(mx, v[j]);
    }
    float s = 0.0f;
    for (int j = 0; j < C; ++j) s += expf(v[j] - mx);
    float ls = logf(s);
    for (int j = 0; j < C; ++j) out[(size_t)row * C + j] = v[j] - mx - ls;
}

// ---------------------------------------------------------------------------
extern "C" void kernel_launch(void* const* d_in, const int* in_sizes, int n_in,
                              void* d_out, int out_size, void* d_ws, size_t ws_size,
                              hipStream_t stream) {
    const float* x   = (const float*)d_in[0];
    const int*   src = (const int*)d_in[1];
    const int*   dst = (const int*)d_in[2];
    const float* W1  = (const float*)d_in[3];
    const float* b1  = (const float*)d_in[4];
    const float* g1  = (const float*)d_in[5];
    const float* be1 = (const float*)d_in[6];
    const float* W2  = (const float*)d_in[7];
    const float* b2  = (const float*)d_in[8];
    const float* g2  = (const float*)d_in[9];
    const float* be2 = (const float*)d_in[10];
    const float* W3  = (const float*)d_in[11];
    const float* b3  = (const float*)d_in[12];

    const int D = in_sizes[4];           // 128
    const int N = in_sizes[0] / D;       // 100000
    const int E = in_sizes[1];           // 1600000
    const int C = in_sizes[12];          // 40
    (void)n_in; (void)out_size; (void)ws_size;

    // Workspace carve-up (~155 MB)
    const size_t nd = (size_t)N * 128;
    float* hbuf   = (float*)d_ws;           // N x 128  scaled features
    float* agg    = hbuf + nd;              // N x 128  spmm accumulator
    float* obuf   = agg + nd;               // N x 128  gemm output
    float* degs   = obuf + nd;              // 2N: deg_out, deg_in
    float* dinvs  = degs + 2 * (size_t)N;   // 2N: dinv_out, dinv_in
    float* bnstat = dinvs + 2 * (size_t)N;  // 256: sum[128], ssq[128]
    float* bnsc   = bnstat + 256;           // 256: scale[128], shift[128]

    const int n4       = N * 32;                      // float4 count of N x 128
    const int gridE    = (E + 255) / 256;
    const int gridN    = (N + 255) / 256;
    const int gridN4   = (n4 + 255) / 256;
    const int gridSpmm = (E * 32 + 255) / 256;        // one wave per edge
    const int gridGemm = (N + 127) / 128;
    const float invN   = 1.0f / (float)N;

    // Degrees
    hipMemsetAsync(degs, 0, 2 * (size_t)N * sizeof(float), stream);
    gcn_degree<<<gridE, 256, 0, stream>>>(src, dst, degs, degs + N, E);
    gcn_deginv<<<gridN, 256, 0, stream>>>(degs, degs + N, dinvs, dinvs + N, N);

    // h0 = x * dout^{-1/2}
    gcn_scale<<<gridN4, 256, 0, stream>>>(x, dinvs, hbuf, n4);

    // ---- Layer 1 ----
    hipMemsetAsync(agg, 0, nd * sizeof(float), stream);
    gcn_spmm<<<gridSpmm, 256, 0, stream>>>(hbuf, src, dst, agg, E);
    hipMemsetAsync(bnstat, 0, 256 * sizeof(float), stream);
    gcn_gemm<128, true><<<gridGemm, 256, 0, stream>>>(
        agg, dinvs + N, W1, b1, obuf, 128, 128, bnstat, bnstat + 128, N);
    gcn_bnfinal<<<1, 128, 0, stream>>>(bnstat, bnstat + 128, g1, be1, bnsc, bnsc + 128, invN);
    gcn_bnapply<<<gridN4, 256, 0, stream>>>(obuf, bnsc, bnsc + 128, dinvs, hbuf, n4);

    // ---- Layer 2 ----
    hipMemsetAsync(agg, 0, nd * sizeof(float), stream);
    gcn_spmm<<<gridSpmm, 256, 0, stream>>>(hbuf, src, dst, agg, E);
    hipMemsetAsync(bnstat, 0, 256 * sizeof(float), stream);
    gcn_gemm<128, true><<<gridGemm, 256, 0, stream>>>(
        agg, dinvs + N, W2, b2, obuf, 128, 128, bnstat, bnstat + 128, N);
    gcn_bnfinal<<<1, 128, 0, stream>>>(bnstat, bnstat + 128, g2, be2, bnsc, bnsc + 128, invN);
    gcn_bnapply<<<gridN4, 256, 0, stream>>>(obuf, bnsc, bnsc + 128, dinvs, hbuf, n4);

    // ---- Layer 3 + log_softmax ----
    hipMemsetAsync(agg, 0, nd * sizeof(float), stream);
    gcn_spmm<<<gridSpmm, 256, 0, stream>>>(hbuf, src, dst, agg, E);
    gcn_gemm<48, false><<<gridGemm, 256, 0, stream>>>(
        agg, dinvs + N, W3, b3, (float*)d_out, C, C, nullptr, nullptr, N);
    gcn_logsoftmax<<<gridN, 256, 0, stream>>>((float*)d_out, N, C);
}